// MPAINNPrediction_48120813585085
// MI455X (gfx1250) — compile-verified
//
#include <hip/hip_runtime.h>

// ---------------------------------------------------------------------------
// Problem constants (from reference): N=1e6 rows, D=64 feats, slice=[48:64],
// HID=16, G=8192 graphs. F_hat is identically zero (energy ignores pos).
// ---------------------------------------------------------------------------
#define NROWS 1000000
#define DFEAT 64
#define SPLIT 48
#define HID   16
#define NSEG  8192

typedef float v2f __attribute__((ext_vector_type(2)));
typedef float v4f __attribute__((ext_vector_type(4)));
typedef float v8f __attribute__((ext_vector_type(8)));

// ---------------------------------------------------------------------------
// Kernel 1: zero the whole output (E accumulator region + F_hat zeros).
// out_size (3,008,192) is divisible by 4 -> float4 grid-stride stores.
// ---------------------------------------------------------------------------
__global__ __launch_bounds__(256) void zero_fill_kernel(float* __restrict__ out,
                                                        int n4) {
  v4f z = {0.f, 0.f, 0.f, 0.f};
  int i = blockIdx.x * blockDim.x + threadIdx.x;
  int stride = gridDim.x * blockDim.x;
  v4f* o = (v4f*)out;
  for (; i < n4; i += stride) o[i] = z;
}

// ---------------------------------------------------------------------------
// Kernel 2: per-row tiny MLP via V_WMMA_F32_16X16X4_F32 + segment-sum.
// One wave32 handles a tile of 16 consecutive rows:
//   A (16x16 slice of x) @ B (w1, 16x16) done as 4 chained K=4 fp32 WMMAs,
//   bias b1 folded into the initial accumulator.
// Then silu + column-scale by w2 (hardware v_rcp_f32 for sigmoid), shfl_xor
// reduce across the 16-lane column groups, atomicAdd into E[batch[row]].
// ---------------------------------------------------------------------------
__global__ __launch_bounds__(256) void energy_kernel(
    const float* __restrict__ x,      // [NROWS, 64]
    const int*   __restrict__ batch,  // [NROWS], sorted
    const float* __restrict__ w1,     // [16,16] row-major
    const float* __restrict__ b1,     // [16]
    const float* __restrict__ w2,     // [16]
    const float* __restrict__ b2,     // [1]
    float* __restrict__ E) {          // [NSEG]
  const int lane = threadIdx.x & 31;
  const int wave = threadIdx.x >> 5;
  const int tile = blockIdx.x * 8 + wave;          // 8 waves / block
  const int numTiles = NROWS / 16;                  // 62,500 exactly
  if (tile >= numTiles) return;                     // wave-uniform guard
  const int row0 = tile * 16;

  const int n  = lane & 15;   // row M for A loads; column N for B/C/D
  const int hi = lane >> 4;   // half-wave select (K sub-pair / M+8 rows)

  // ---- Load A tile: lane holds row M=n, K pair per WMMA step.
  //      16x4 f32 A layout: VGPR0/1 = {K, K+1}; low half K=4kk, high K=4kk+2.
  v2f a[4];
  const float* xrow = x + (size_t)(row0 + n) * DFEAT + SPLIT;
#pragma unroll
  for (int kk = 0; kk < 4; ++kk) {
    const int k = 4 * kk + 2 * hi;
    a[kk].x = xrow[k];
    a[kk].y = xrow[k + 1];
  }

  // ---- Load B tile: w1 rows K (striped across lanes as columns N=n).
  v2f b[4];
#pragma unroll
  for (int kk = 0; kk < 4; ++kk) {
    const int k = 4 * kk + 2 * hi;
    b[kk].x = w1[(k + 0) * HID + n];
    b[kk].y = w1[(k + 1) * HID + n];
  }

  // ---- Accumulator pre-loaded with bias b1 (per column N).
  const float bias = b1[n];
  v8f c;
#pragma unroll
  for (int r = 0; r < 8; ++r) c[r] = bias;

  // ---- 16x16x16 fp32 GEMM = 4 x V_WMMA_F32_16X16X4_F32.
#pragma unroll
  for (int kk = 0; kk < 4; ++kk) {
    c = __builtin_amdgcn_wmma_f32_16x16x4_f32(
        /*neg_a=*/false, a[kk], /*neg_b=*/false, b[kk],
        /*c_mod=*/(short)0, c, /*reuse_a=*/false, /*reuse_b=*/false);
  }

  // ---- silu(h) * w2[N] per element. C/D layout: c[r] = h[M = r+8*hi][N = n].
  //      sigmoid via hardware v_rcp_f32 (single TRANS op) instead of the
  //      IEEE-division expansion: silu(z) = z * rcp(1 + exp(-z)).
  const float w2n = w2[n];
  float partial[8];
#pragma unroll
  for (int r = 0; r < 8; ++r) {
    const float z = c[r];
    const float sig = __builtin_amdgcn_rcpf(1.0f + __expf(-z));
    partial[r] = z * sig * w2n;
  }

  // ---- Reduce over N: xor-shuffle within each 16-lane column group.
#pragma unroll
  for (int r = 0; r < 8; ++r) {
#pragma unroll
    for (int m = 8; m >= 1; m >>= 1) {
      partial[r] += __shfl_xor(partial[r], m, 16);
    }
  }

  // ---- Lanes n<8 own row M = n + 8*hi; segment-sum into E.
  if (n < 8) {
    const int M = n + 8 * hi;
    const float e = partial[n] + b2[0];
    atomicAdd(&E[batch[row0 + M]], e);
  }
}

// ---------------------------------------------------------------------------
// Host launcher. Inputs (setup_inputs order):
//   0:x  1:pos  2:batch  3:num_graphs  4:w1  5:b1  6:w2  7:b2
// Output: out[0:8192]=E_hat, out[8192:3008192]=F_hat (all zeros).
// ---------------------------------------------------------------------------
extern "C" void kernel_launch(void* const* d_in, const int* in_sizes, int n_in,
                              void* d_out, int out_size, void* d_ws, size_t ws_size,
                              hipStream_t stream) {
  const float* x     = (const float*)d_in[0];
  const int*   batch = (const int*)  d_in[2];
  const float* w1    = (const float*)d_in[4];
  const float* b1    = (const float*)d_in[5];
  const float* w2    = (const float*)d_in[6];
  const float* b2    = (const float*)d_in[7];
  float* out = (float*)d_out;

  // 1) Zero E accumulators + F_hat region (12 MB of stores, float4-wide).
  const int n4 = out_size / 4;  // 3,008,192 / 4 = 752,048
  zero_fill_kernel<<<2048, 256, 0, stream>>>(out, n4);

  // 2) WMMA MLP + segment sum. One wave per 16-row tile, 8 waves per block.
  const int numTiles = NROWS / 16;            // 62,500
  const int blocks = (numTiles + 7) / 8;      // 7,813
  energy_kernel<<<blocks, 256, 0, stream>>>(x, batch, w1, b1, w2, b2, out);
}